// SwitchFocusedLossAdditive3Class_80221399155361
// MI455X (gfx1250) — compile-verified
//
#include <hip/hip_runtime.h>
#include <stdint.h>

#define TOL   5
#define NROWS 512
#define SEQ   16384
#define BPR   16                      // blocks per row
#define CHUNK (SEQ / BPR)             // 1024
#define NTHR  256                     // 8 wave32s
#define ELEMS (CHUNK / NTHR)          // 4
#define HALO  (2 * TOL)

typedef float v2f __attribute__((ext_vector_type(2)));
typedef float v8f __attribute__((ext_vector_type(8)));

__device__ __forceinline__ void async_wait0() {
#if __has_builtin(__builtin_amdgcn_s_wait_asynccnt)
  __builtin_amdgcn_s_wait_asynccnt(0);
#else
  asm volatile("s_wait_asynccnt 0x0" ::: "memory");
#endif
}

__global__ __launch_bounds__(NTHR) void switch_loss_main(
    const float* __restrict__ logits,
    const int*   __restrict__ labels,
    float4*      __restrict__ part)
{
  __shared__ int           s_lab[CHUNK];
  __shared__ unsigned char s_mask[CHUNK + HALO];   // bit0 = true_sw, bit1 = pred_sw
  __shared__ float         s_red[(NTHR / 32) * 4];

  const int  tid     = threadIdx.x;
  const int  row     = blockIdx.y;
  const int  c0      = blockIdx.x * CHUNK;
  const long rowBase = (long)row * SEQ;

  // ---- async (ASYNCcnt) gather of this chunk's labels into LDS ------------
#pragma unroll
  for (int k = 0; k < ELEMS; ++k) {
    const int i = k * NTHR + tid;
    unsigned ldsOff = (unsigned)(uintptr_t)(&s_lab[i]);
    unsigned long long ga =
        (unsigned long long)(uintptr_t)(labels + rowBase + c0 + i);
    asm volatile("global_load_async_to_lds_b32 %0, %1, off"
                 :: "v"(ldsOff), "v"(ga) : "memory");
  }

  // ---- halo masks (left TOL, right TOL), clipped at row edges -------------
  if (tid < HALO) {
    const int left = (tid < TOL);
    const int pos  = left ? (c0 - TOL + tid) : (c0 + CHUNK + (tid - TOL));
    const int li   = left ? tid : (TOL + CHUNK + (tid - TOL));
    unsigned char m = 0;
    if (pos >= 0 && pos < SEQ) {
      const long e = rowBase + pos;
      const float* lp = logits + e * 3;
      const float l0 = lp[0], l1 = lp[1], l2 = lp[2];
      const int predsw = !((l0 >= l1) && (l0 >= l2));   // argmax >= 1
      const int truesw = (labels[e] >= 1);
      m = (unsigned char)((predsw << 1) | truesw);
    }
    s_mask[li] = m;
  }

  // ---- per-element logits: logsumexp + argmax (overlaps async loads) ------
  float l0r[ELEMS], l1r[ELEMS], l2r[ELEMS], lser[ELEMS];
  int predr[ELEMS];
#pragma unroll
  for (int k = 0; k < ELEMS; ++k) {
    const int  i = k * NTHR + tid;
    const long e = rowBase + c0 + i;
    const float* lp = logits + e * 3;
    const float l0 = lp[0], l1 = lp[1], l2 = lp[2];
    l0r[k] = l0; l1r[k] = l1; l2r[k] = l2;
    const float mx = fmaxf(l0, fmaxf(l1, l2));
    lser[k] = mx + logf(expf(l0 - mx) + expf(l1 - mx) + expf(l2 - mx));
    predr[k] = !((l0 >= l1) && (l0 >= l2));
  }

  async_wait0();   // our own s_lab slots now valid (per-wave ASYNCcnt)

  float basek[ELEMS];
  int   truek[ELEMS], validk[ELEMS];
  float validAcc = 0.f;
  int   anyTrue  = 0;
#pragma unroll
  for (int k = 0; k < ELEMS; ++k) {
    const int i     = k * NTHR + tid;
    const int lab   = s_lab[i];
    const int valid = (lab != -100);
    const int safe  = valid ? lab : 0;
    const int truesw = (lab >= 1);                    // -100 < 1 -> false
    s_mask[TOL + i] = (unsigned char)((predr[k] << 1) | truesw);
    const float pick = (safe == 0 ? l0r[k] : (safe == 1 ? l1r[k] : l2r[k])) - lser[k];
    const float w    = (safe == 0) ? 0.1f : 5.0f;     // class weights {0.1,5,5}
    basek[k]  = valid ? (-w * pick) : 0.f;
    truek[k]  = truesw;
    validk[k] = valid;
    anyTrue  |= truesw;
    validAcc += valid ? 1.f : 0.f;
  }

  __syncthreads();  // all masks (halo + chunk) visible to the whole block

  float mainAcc = 0.f, penAcc = 0.f;
#pragma unroll
  for (int k = 0; k < ELEMS; ++k) {
    const int i = k * NTHR + tid;
    unsigned orv = 0;
#pragma unroll
    for (int j = 0; j <= HALO; ++j) orv |= (unsigned)s_mask[i + j];
    const int predNear = (int)((orv >> 1) & 1u);
    const int trueNear = (int)(orv & 1u);
    float v = basek[k];
    if (truek[k] & predNear) v -= 2.0f;                       // proximity reward
    mainAcc += v;
    if (validk[k] && predr[k] && !trueNear) penAcc += 1.5f;   // far penalty (row-gated later)
  }

  // ---- block reduction (wave32 shuffles + LDS across 8 waves) -------------
#pragma unroll
  for (int o = 16; o > 0; o >>= 1) {
    mainAcc  += __shfl_down(mainAcc,  o, 32);
    penAcc   += __shfl_down(penAcc,   o, 32);
    validAcc += __shfl_down(validAcc, o, 32);
    anyTrue  |= __shfl_down(anyTrue,  o, 32);
  }
  const int wv = tid >> 5;
  if ((tid & 31) == 0) {
    s_red[wv * 4 + 0] = mainAcc;
    s_red[wv * 4 + 1] = penAcc;
    s_red[wv * 4 + 2] = validAcc;
    s_red[wv * 4 + 3] = anyTrue ? 1.f : 0.f;
  }
  __syncthreads();
  if (tid == 0) {
    float m = 0.f, p = 0.f, vv = 0.f, h = 0.f;
    for (int q = 0; q < NTHR / 32; ++q) {
      m  += s_red[q * 4 + 0];
      p  += s_red[q * 4 + 1];
      vv += s_red[q * 4 + 2];
      h   = fmaxf(h, s_red[q * 4 + 3]);
    }
    part[(size_t)row * BPR + blockIdx.x] = make_float4(m, p, vv, h);
  }
}

// Single-wave finisher: gate penalties per row, WMMA-reduce 512 row totals.
__global__ __launch_bounds__(32) void switch_loss_final(
    const float4* __restrict__ part, float* __restrict__ out)
{
  const int lane = threadIdx.x;          // one full wave32, EXEC all ones
  float rt[NROWS / 32];                  // 16 gated row totals per lane
  float validTot = 0.f;
#pragma unroll
  for (int j = 0; j < NROWS / 32; ++j) {
    const int r = j * 32 + lane;
    float m = 0.f, p = 0.f, vv = 0.f, h = 0.f;
    for (int b = 0; b < BPR; ++b) {
      const float4 q = part[(size_t)r * BPR + b];
      m += q.x; p += q.y; vv += q.z; h = fmaxf(h, q.w);
    }
    rt[j]     = m + (h > 0.5f ? p : 0.f);   // has_true gate per row
    validTot += vv;
  }

  // D = ones(16x4) x B + C  =>  D[m][n] = colsum(B)[n]; summing D's columns
  // equals the sum of ALL 64 B elements irrespective of B's lane layout.
  const v2f aOnes = {1.f, 1.f};
  v8f c = {};
#pragma unroll
  for (int q = 0; q < 8; ++q) {
    v2f b; b[0] = rt[2 * q]; b[1] = rt[2 * q + 1];
    c = __builtin_amdgcn_wmma_f32_16x16x4_f32(false, aOnes, false, b,
                                              (short)0, c, false, false);
  }
  float colsum = c[0];                   // col sum for N = lane%16 (replicated over M)
#pragma unroll
  for (int o = 8; o > 0; o >>= 1) colsum += __shfl_xor(colsum, o, 16);
#pragma unroll
  for (int o = 16; o > 0; o >>= 1) validTot += __shfl_xor(validTot, o, 32);
  if (lane == 0) out[0] = colsum / validTot;
}

extern "C" void kernel_launch(void* const* d_in, const int* in_sizes, int n_in,
                              void* d_out, int out_size, void* d_ws, size_t ws_size,
                              hipStream_t stream) {
  const float* logits = (const float*)d_in[0];
  const int*   labels = (const int*)d_in[1];
  float*       out    = (float*)d_out;
  float4*      part   = (float4*)d_ws;   // NROWS*BPR*16B = 128 KiB, fully rewritten each call

  dim3 grid(BPR, NROWS);
  switch_loss_main<<<grid, NTHR, 0, stream>>>(logits, labels, part);
  switch_loss_final<<<1, 32, 0, stream>>>(part, out);

  (void)in_sizes; (void)n_in; (void)out_size; (void)ws_size;
}